// batch_Q_approx_51135880626198
// MI455X (gfx1250) — compile-verified
//
#include <hip/hip_runtime.h>
#include <hip/hip_bf16.h>

// Batched per-net MLP forward: 4096 independent nets, H=256, 3 actions.
// h1 = relu(w1+b1); h2 = relu(h1 @ W2 + b2); out = h2 @ W3 + b3
// Bandwidth-bound (w2 = 1 GiB streamed once). Layer-2 matvec is done on the
// matrix pipe via V_WMMA_F32_16X16X4_F32 with a padded A (only row M=0 live).

typedef float v2f __attribute__((ext_vector_type(2)));
typedef float v8f __attribute__((ext_vector_type(8)));

constexpr int N_NETS = 4096;
constexpr int H      = 256;
constexpr int NACT   = 3;

__global__ __launch_bounds__(256) void qnet_fwd_kernel(
    const float* __restrict__ w1, const float* __restrict__ b1,
    const float* __restrict__ w2, const float* __restrict__ b2,
    const float* __restrict__ w3, const float* __restrict__ b3,
    float* __restrict__ out)
{
    // h1 padded with a zero region so A-fragment builds are branchless ds_load_b64:
    // lanes 0/16 read live h1 values, all other lanes read zeros at offset >= H.
    __shared__ __align__(16) float h1p[2 * H + 8];
    __shared__ float h2s[H];

    const int n    = blockIdx.x;            // net id
    const int t    = threadIdx.x;           // 0..255
    const int lane = t & 31;                // wave32 lane
    const int wave = t >> 5;                // 0..7

    // ---------------- Layer 1: h1 = relu(w1 + b1) ----------------
    {
        float v = w1[n * H + t] + b1[n * H + t];
        h1p[t]     = fmaxf(v, 0.0f);
        h1p[H + t] = 0.0f;                  // zero pad region
    }
    if (t < 8) h1p[2 * H + t] = 0.0f;
    __syncthreads();

    // ---------------- Layer 2: h2 = relu(h1 @ W2 + b2) via WMMA ----------------
    // Wave w owns output columns [32w, 32w+32) as two 16-wide tiles.
    //
    // A (16x4 f32, only M=0 row nonzero):
    //   VGPR0: lanes0-15 K=0, lanes16-31 K=2 ; VGPR1: K=1 / K=3  (M = lane%16)
    //   -> lane0 needs h1[k],h1[k+1]; lane16 needs h1[k+2],h1[k+3]; others 0.
    // B (4x16 f32, rows striped across lanes per VGPR):
    //   VGPR0: lanes0-15 row K=0, lanes16-31 row K=2 ; VGPR1: rows K=1/K=3.
    //   Row-major w2 => each B VGPR is two contiguous 64B row segments: coalesced.
    const int c0   = wave * 32;
    const int aoff = (lane == 0) ? 0 : ((lane == 16) ? 2 : H);  // zero pad for most lanes
    const int rowb = (lane < 16) ? 0 : 2;
    const int col  = c0 + (lane & 15);

    const float* __restrict__ bp = w2 + (size_t)n * H * H + (size_t)rowb * H + col;

    v8f acc0 = {};
    v8f acc1 = {};

    #pragma unroll 4
    for (int k = 0; k < H; k += 4) {
        v2f a = *(const v2f*)(h1p + k + aoff);   // A fragment (one ds_load_b64)

        v2f b0, b1v;
        b0.x  = __builtin_nontemporal_load(bp);            // row k+rowb,   col
        b0.y  = __builtin_nontemporal_load(bp + H);        // row k+rowb+1, col
        b1v.x = __builtin_nontemporal_load(bp + 16);       // tile 1
        b1v.y = __builtin_nontemporal_load(bp + H + 16);

        acc0 = __builtin_amdgcn_wmma_f32_16x16x4_f32(
                   false, a, false, b0,  (short)0, acc0, false, false);
        acc1 = __builtin_amdgcn_wmma_f32_16x16x4_f32(
                   false, a, false, b1v, (short)0, acc1, false, false);

        bp += 4 * H;
    }

    // D row M=0 lives in accumulator VGPR0 on lanes 0-15 (N = lane).
    if (lane < 16) {
        const int c = c0 + lane;
        h2s[c]      = fmaxf(acc0[0] + b2[n * H + c],      0.0f);
        h2s[c + 16] = fmaxf(acc1[0] + b2[n * H + c + 16], 0.0f);
    }
    __syncthreads();

    // ---------------- Layer 3: out = h2 @ W3 + b3 (256x3, tiny) ----------------
    if (wave < NACT) {
        const int a = wave;
        const float* __restrict__ w3n = w3 + (size_t)n * H * NACT;
        float s = 0.0f;
        #pragma unroll
        for (int h = lane; h < H; h += 32)
            s += h2s[h] * w3n[h * NACT + a];
        #pragma unroll
        for (int off = 16; off > 0; off >>= 1)
            s += __shfl_down(s, off, 32);
        if (lane == 0)
            out[n * NACT + a] = s + b3[n * NACT + a];
    }
}

extern "C" void kernel_launch(void* const* d_in, const int* in_sizes, int n_in,
                              void* d_out, int out_size, void* d_ws, size_t ws_size,
                              hipStream_t stream) {
    (void)in_sizes; (void)n_in; (void)out_size; (void)d_ws; (void)ws_size;
    // setup_inputs order: a(int64, unused), w1, b1, w2, b2, w3, b3
    const float* w1 = (const float*)d_in[1];
    const float* b1 = (const float*)d_in[2];
    const float* w2 = (const float*)d_in[3];
    const float* b2 = (const float*)d_in[4];
    const float* w3 = (const float*)d_in[5];
    const float* b3 = (const float*)d_in[6];
    float* out = (float*)d_out;

    qnet_fwd_kernel<<<N_NETS, 256, 0, stream>>>(w1, b1, w2, b2, w3, b3, out);
}